// SoftCrossEntropyLoss_62199716381128
// MI455X (gfx1250) — compile-verified
//
#include <hip/hip_runtime.h>

typedef __attribute__((ext_vector_type(2))) float v2f;
typedef __attribute__((ext_vector_type(8))) float v8f;

namespace {
constexpr int kK = 512;            // row length (fixed by reference)
constexpr int kBlocks = 2048;      // partial-sum count (fits 8 KB of d_ws)
constexpr int kThreads = 256;      // 8 waves per block (wave32)
constexpr int kWaves = kThreads / 32;
}

// Kernel 1: one row per wave per iteration. Row (512 f32 = 2 KB) lives in 16
// VGPRs/lane so logits are read exactly once from HBM. Streams ~1.07 GB total.
__global__ __launch_bounds__(kThreads) void sce_rows_kernel(
    const float* __restrict__ logits,
    const float* __restrict__ target,
    const float* __restrict__ weight,
    float* __restrict__ partial,
    int n_rows) {
  const int lane  = threadIdx.x & 31;
  const int wave  = threadIdx.x >> 5;
  const int gwave = blockIdx.x * kWaves + wave;
  const int nwaves = gridDim.x * kWaves;

  // weight[K] cached in registers once per wave (16 floats/lane, coalesced)
  float4 w[4];
#pragma unroll
  for (int c = 0; c < 4; ++c)
    w[c] = *reinterpret_cast<const float4*>(weight + 4 * (lane + 32 * c));

  float wave_acc = 0.0f;

  for (int row = gwave; row < n_rows; row += nwaves) {
    const float* xrow = logits + (size_t)row * kK;
    const float* trow = target + (size_t)row * kK;

    // L2 streaming prefetch of this wave's next row (emits global_prefetch_b8)
    if (row + nwaves < n_rows) {
      const float* xn = logits + (size_t)(row + nwaves) * kK;
      const float* tn = target + (size_t)(row + nwaves) * kK;
#pragma unroll
      for (int c = 0; c < 4; ++c) {
        __builtin_prefetch(xn + 4 * (lane + 32 * c), 0, 0);
        __builtin_prefetch(tn + 4 * (lane + 32 * c), 0, 0);
      }
    }

    float4 x[4], t[4];
#pragma unroll
    for (int c = 0; c < 4; ++c)
      x[c] = *reinterpret_cast<const float4*>(xrow + 4 * (lane + 32 * c));
#pragma unroll
    for (int c = 0; c < 4; ++c)
      t[c] = *reinterpret_cast<const float4*>(trow + 4 * (lane + 32 * c));

    // row max (lane-local then wave32 xor tree)
    float m = -__builtin_inff();
#pragma unroll
    for (int c = 0; c < 4; ++c)
      m = fmaxf(m, fmaxf(fmaxf(x[c].x, x[c].y), fmaxf(x[c].z, x[c].w)));
#pragma unroll
    for (int off = 16; off > 0; off >>= 1)
      m = fmaxf(m, __shfl_xor(m, off, 32));

    // sum exp(x - m)
    float s = 0.0f;
#pragma unroll
    for (int c = 0; c < 4; ++c)
      s += __expf(x[c].x - m) + __expf(x[c].y - m) +
           __expf(x[c].z - m) + __expf(x[c].w - m);
#pragma unroll
    for (int off = 16; off > 0; off >>= 1)
      s += __shfl_xor(s, off, 32);

    const float logZ = m + __logf(s);

    // weighted dot with log-softmax, reusing x[] from registers (no re-read)
    float acc = 0.0f;
#pragma unroll
    for (int c = 0; c < 4; ++c) {
      acc += t[c].x * w[c].x * (x[c].x - logZ);
      acc += t[c].y * w[c].y * (x[c].y - logZ);
      acc += t[c].z * w[c].z * (x[c].z - logZ);
      acc += t[c].w * w[c].w * (x[c].w - logZ);
    }
#pragma unroll
    for (int off = 16; off > 0; off >>= 1)
      acc += __shfl_xor(acc, off, 32);

    wave_acc -= acc;  // loss is negated sum; acc is identical in every lane
  }

  __shared__ float lds_wsum[kWaves];
  if (lane == 0) lds_wsum[wave] = wave_acc;
  __syncthreads();
  if (threadIdx.x == 0) {
    float b = 0.0f;
#pragma unroll
    for (int i = 0; i < kWaves; ++i) b += lds_wsum[i];
    partial[blockIdx.x] = b;
  }
}

// Kernel 2: single wave reduces the 2048 partials with V_WMMA_F32_16X16X4_F32.
// B = all-ones => C[i][j] = running row-sum of A; 64 values folded per WMMA.
// C column 0 lives in lane 0 (M=0..7) and lane 16 (M=8..15) per the ISA
// 16x16 f32 C/D layout, so one 8-way add + one xor-16 shuffle finishes it.
__global__ __launch_bounds__(32) void sce_reduce_kernel(
    const float* __restrict__ partial, float* __restrict__ out, float inv_n) {
  const int lane = threadIdx.x;  // exactly one wave32, EXEC all ones
  v8f c = {0.f, 0.f, 0.f, 0.f, 0.f, 0.f, 0.f, 0.f};
  const v2f bones = {1.0f, 1.0f};
  for (int base = 0; base < kBlocks; base += 64) {
    v2f a;
    a.x = partial[base + 2 * lane + 0];
    a.y = partial[base + 2 * lane + 1];
    // 8 args: (neg_a, A, neg_b, B, c_mod, C, reuse_a, reuse_b)
    c = __builtin_amdgcn_wmma_f32_16x16x4_f32(false, a, false, bones,
                                              (short)0, c, false, false);
  }
  float lsum = c[0] + c[1] + c[2] + c[3] + c[4] + c[5] + c[6] + c[7];
  const float total = lsum + __shfl_xor(lsum, 16, 32);
  if (lane == 0) out[0] = total * inv_n;
}

extern "C" void kernel_launch(void* const* d_in, const int* in_sizes, int n_in,
                              void* d_out, int out_size, void* d_ws, size_t ws_size,
                              hipStream_t stream) {
  const float* logits = (const float*)d_in[0];
  const float* target = (const float*)d_in[1];
  const float* weight = (const float*)d_in[2];
  float* out = (float*)d_out;
  float* partial = (float*)d_ws;  // kBlocks floats = 8 KB
  const int n_rows = in_sizes[0] / kK;

  sce_rows_kernel<<<kBlocks, kThreads, 0, stream>>>(logits, target, weight,
                                                    partial, n_rows);
  sce_reduce_kernel<<<1, 32, 0, stream>>>(partial, out, 1.0f / (float)n_rows);
}